// ParatopeAwareReadout_27582279975196
// MI455X (gfx1250) — compile-verified
//
#include <hip/hip_runtime.h>
#include <hip/hip_bf16.h>

#define N_NODES 1048576
#define DIM     256
#define NB      1024
#define BETA    1.0f
#define GAMMA   0.5f
#define TILE    64
#define DIMP    264   // padded LDS row pitch in floats (1056 B, 16B aligned)

typedef __attribute__((ext_vector_type(2))) float v2f;
typedef __attribute__((ext_vector_type(4))) float v4f;
typedef __attribute__((ext_vector_type(8))) float v8f;
typedef int v4i __attribute__((vector_size(16)));   // matches builtin param type

#if defined(__has_builtin)
#  if __has_builtin(__builtin_amdgcn_global_load_async_to_lds_b128)
#    define HAVE_ASYNC_LDS 1
#  endif
#endif

__device__ __forceinline__ void wait_async_zero() {
#if defined(HAVE_ASYNC_LDS)
#  if __has_builtin(__builtin_amdgcn_s_wait_asynccnt)
  __builtin_amdgcn_s_wait_asynccnt(0);
#  else
  asm volatile("s_wait_asynccnt 0" ::: "memory");
#  endif
#endif
}

#if defined(HAVE_ASYNC_LDS)
__device__ __forceinline__ void issue_tile_load_async(
    const float* __restrict__ h, float* buf, int s0, int base, int t) {
  // 64 rows x 64 float4 = 4096 b128 transfers; 256 threads x 16 each.
  #pragma unroll
  for (int j = 0; j < 16; ++j) {
    int lin = t + 256 * j;
    int row = lin >> 6;
    int c4  = lin & 63;
    long grow = (long)s0 + base + row;
    if (grow > (long)N_NODES - 1) grow = (long)N_NODES - 1;  // clamp: finite data, w=0 later
    const float* gp = h + grow * DIM + c4 * 4;
    float* lp = buf + row * DIMP + c4 * 4;
    __builtin_amdgcn_global_load_async_to_lds_b128((v4i*)gp, (v4i*)lp, 0, 0);
  }
}
#endif

__global__ void seg_bounds_kernel(const int* __restrict__ batch, int* __restrict__ start) {
  int b = blockIdx.x * blockDim.x + threadIdx.x;
  if (b > NB) return;
  if (b == NB) { start[b] = N_NODES; return; }
  int lo = 0, hi = N_NODES;                  // lower_bound(batch, b) on sorted batch
  while (lo < hi) { int mid = (lo + hi) >> 1; if (batch[mid] < b) lo = mid + 1; else hi = mid; }
  start[b] = lo;
}

__global__ __launch_bounds__(256) void readout_kernel(
    const float* __restrict__ h, const unsigned char* __restrict__ nmask,
    const float* __restrict__ pp, const float* __restrict__ sa,
    const float* __restrict__ q, const int* __restrict__ start,
    float* __restrict__ out)
{
  __shared__ float buf[2][TILE * DIMP];
  __shared__ float qs[DIM];
  __shared__ float lg[TILE];
  __shared__ float wgt[TILE];
  __shared__ float m_sh, f_sh, s_sh;

  const int b    = blockIdx.x;
  const int t    = threadIdx.x;
  const int lane = t & 31;
  const int wv   = t >> 5;

  const int s0 = start[b];
  const int n  = start[b + 1] - s0;

  if (n <= 0) { out[b * DIM + t] = 0.0f; return; }   // empty graph -> zeros

  qs[t] = q[t];
  if (t == 0) { m_sh = -__builtin_inff(); s_sh = 0.0f; f_sh = 1.0f; }

  // WMMA accumulators: 8 waves x 2 column groups of 16 dims = 256 dims
  v8f c0 = {}; v8f c1 = {};
  const int col   = lane & 15;
  const int n0a   = (wv * 2 + 0) * 16 + col;
  const int n0b   = (wv * 2 + 1) * 16 + col;
  const int khalf = (lane < 16) ? 0 : 2;   // A/B 32-bit layout: lanes 16-31 hold K's upper half

  const int ntiles = (n + TILE - 1) / TILE;

#if defined(HAVE_ASYNC_LDS)
  issue_tile_load_async(h, buf[0], s0, 0, t);
#endif

  for (int tt = 0; tt < ntiles; ++tt) {
    float* cur;
#if defined(HAVE_ASYNC_LDS)
    cur = buf[tt & 1];
    wait_async_zero();
    __syncthreads();  // tile visible; prev tile's lg/wgt/buf consumers done
    if (tt + 1 < ntiles) issue_tile_load_async(h, buf[(tt + 1) & 1], s0, (tt + 1) * TILE, t);
#else
    cur = buf[0];
    __syncthreads();
    #pragma unroll
    for (int j = 0; j < 16; ++j) {
      int lin = t + 256 * j;
      int row = lin >> 6;
      int c4  = lin & 63;
      long grow = (long)s0 + tt * TILE + row;
      if (grow > (long)N_NODES - 1) grow = (long)N_NODES - 1;
      *(v4f*)(cur + row * DIMP + c4 * 4) = *(const v4f*)(h + grow * DIM + c4 * 4);
    }
    __syncthreads();
#endif

    const int base = tt * TILE;
    const int cnt  = (n - base < TILE) ? (n - base) : TILE;

    // ---- logits: 4 threads per node, stride-4 dims (bank-conflict free) ----
    {
      const int i = t >> 2, p = t & 3;
      const float* rowp = cur + i * DIMP;
      float partial = 0.0f;
      #pragma unroll 8
      for (int j = 0; j < 64; ++j) {
        int d = p + 4 * j;
        partial += rowp[d] * qs[d];
      }
      partial += __shfl_xor(partial, 1);
      partial += __shfl_xor(partial, 2);
      if (p == 0) {
        float lv = -__builtin_inff();
        if (i < cnt) {
          int g = s0 + base + i;
          if (nmask[g]) lv = partial + BETA * pp[g] + GAMMA * sa[g];
        }
        lg[i] = lv;
      }
    }
    __syncthreads();

    // ---- tile max -> online max/rescale factor ----
    if (wv == 0) {
      float v = fmaxf(lg[lane], lg[lane + 32]);
      #pragma unroll
      for (int o = 16; o >= 1; o >>= 1) v = fmaxf(v, __shfl_xor(v, o));
      if (lane == 0) {
        float mo = m_sh;
        float mn = fmaxf(mo, v);
        float f  = (mn == -__builtin_inff()) ? 1.0f : __expf(mo - mn);
        m_sh = mn; f_sh = f;
      }
    }
    __syncthreads();

    const float f  = f_sh;
    const float mn = m_sh;
    c0 *= f; c1 *= f;
    if (t < TILE) {
      float lv = lg[t];
      wgt[t] = (lv == -__builtin_inff()) ? 0.0f : __expf(lv - mn);
    }
    __syncthreads();

    if (wv == 0) {
      float sv = wgt[lane] + wgt[lane + 32];
      #pragma unroll
      for (int o = 16; o >= 1; o >>= 1) sv += __shfl_xor(sv, o);
      if (lane == 0) s_sh = s_sh * f + sv;
    }

    // ---- WMMA f32 16x16x4 weighted accumulation over the tile ----
    // A[m][k] = w[kb+k] (all rows equal), B[k][n] = h[kb+k][n0+n]
    #pragma unroll
    for (int kb = 0; kb < TILE; kb += 4) {
      v2f a, b0, b1;
      a.x = wgt[kb + khalf];
      a.y = wgt[kb + khalf + 1];
      const float* r0 = cur + (kb + khalf) * DIMP;
      const float* r1 = cur + (kb + khalf + 1) * DIMP;
      b0.x = r0[n0a]; b0.y = r1[n0a];
      b1.x = r0[n0b]; b1.y = r1[n0b];
      c0 = __builtin_amdgcn_wmma_f32_16x16x4_f32(false, a, false, b0, (short)0, c0, false, false);
      c1 = __builtin_amdgcn_wmma_f32_16x16x4_f32(false, a, false, b1, (short)0, c1, false, false);
    }
  }

  __syncthreads();
  const float s   = s_sh;
  const float inv = (s > 0.0f) ? 1.0f / s : 0.0f;  // zero-output for fully-masked graphs
  if (lane < 16) {   // D row M=0 lives in VGPR0 of lanes 0-15
    out[b * DIM + n0a] = c0[0] * inv;
    out[b * DIM + n0b] = c1[0] * inv;
  }
}

extern "C" void kernel_launch(void* const* d_in, const int* in_sizes, int n_in,
                              void* d_out, int out_size, void* d_ws, size_t ws_size,
                              hipStream_t stream) {
  const float*         h     = (const float*)d_in[0];
  const int*           batch = (const int*)d_in[1];
  const unsigned char* nm    = (const unsigned char*)d_in[2];
  const float*         pp    = (const float*)d_in[3];
  const float*         sa    = (const float*)d_in[4];
  const float*         q     = (const float*)d_in[5];
  float* out = (float*)d_out;
  int*   start = (int*)d_ws;  // (NB+1) ints of scratch

  seg_bounds_kernel<<<(NB + 1 + 255) / 256, 256, 0, stream>>>(batch, start);
  readout_kernel<<<NB, 256, 0, stream>>>(h, nm, pp, sa, q, start, out);
}